// MAGNN_metapath_specific_53008486367319
// MI455X (gfx1250) — compile-verified
//
#include <hip/hip_runtime.h>
#include <hip/hip_bf16.h>

#define OUT_DIM   64
#define D2        32
#define NUM_HEADS 8
#define ALPHA_C   0.01f
#define KDIM      128   // [center(64) | hidden(64)]

typedef __attribute__((ext_vector_type(2))) float v2f;
typedef __attribute__((ext_vector_type(8))) float v8f;

// ---- ordered-int encoding for f32 atomic max ----
__device__ __forceinline__ int f2ord(float f) {
    int i = __float_as_int(f);
    return (i >= 0) ? i : (i ^ 0x7fffffff);
}
__device__ __forceinline__ float ord2f(int i) {
    return __int_as_float((i >= 0) ? i : (i ^ 0x7fffffff));
}

__device__ __forceinline__ float2 cmul(float2 a, float2 b) {
    float2 r;
    r.x = a.x * b.x - a.y * b.y;
    r.y = a.x * b.y + a.y * b.x;
    return r;
}

// ---------------------------------------------------------------------------
// Kernel 0: init output, segment max (ordered INT_MIN), segment sum (0)
// ---------------------------------------------------------------------------
__global__ void init_kernel(float* __restrict__ out, int* __restrict__ amaxOrd,
                            float* __restrict__ asum, int N) {
    long long i = (long long)blockIdx.x * blockDim.x + threadIdx.x;
    long long nh = (long long)N * NUM_HEADS;
    if (i < nh) { amaxOrd[i] = (int)0x80000000; asum[i] = 0.0f; }
    if (i < nh * OUT_DIM) out[i] = 0.0f;
}

// ---------------------------------------------------------------------------
// Kernel 1: build frv (L=4, 32 complex) from r_vec. ETYPES=[0,2,1] hardcoded.
// fr[3]=1; fr[2]=conj(rv0); fr[1]=fr[2]*rv1; fr[0]=fr[1]*rv0
// ---------------------------------------------------------------------------
__global__ void frv_kernel(const float* __restrict__ r_vec, float* __restrict__ frv) {
    int d = threadIdx.x;  // 0..31 complex dim
    float2 r0 = *(const float2*)(r_vec + d * 2);
    float2 r1 = *(const float2*)(r_vec + 64 + d * 2);
    float n0 = 1.0f / sqrtf(r0.x * r0.x + r0.y * r0.y);
    float n1 = 1.0f / sqrtf(r1.x * r1.x + r1.y * r1.y);
    r0.x *= n0; r0.y *= n0;
    r1.x *= n1; r1.y *= n1;
    float2 fr3 = make_float2(1.0f, 0.0f);
    float2 fr2 = make_float2(r0.x, -r0.y);       // 1 * conj(rv[0])
    float2 fr1 = cmul(fr2, r1);                  // * rv[1]
    float2 fr0 = cmul(fr1, r0);                  // * rv[0]
    *(float2*)(frv + 0 * 64 + 2 * d) = fr0;
    *(float2*)(frv + 1 * 64 + 2 * d) = fr1;
    *(float2*)(frv + 2 * 64 + 2 * d) = fr2;
    *(float2*)(frv + 3 * 64 + 2 * d) = fr3;
}

// ---------------------------------------------------------------------------
// Kernel 2: hidden[e,:] = mean_l rotate(features[idx[e,l]], frv[l])
// One wave32 per edge; lane = complex dim (float2).
// ---------------------------------------------------------------------------
__global__ void hidden_kernel(const float* __restrict__ features,
                              const int* __restrict__ emi,
                              const float* __restrict__ frv,
                              float* __restrict__ hid, int E) {
    int lane = threadIdx.x & 31;
    int e = blockIdx.x * 8 + (threadIdx.x >> 5);
    if (e >= E) return;
    int4 iv = *(const int4*)(emi + (size_t)e * 4);
    int idx[4] = {iv.x, iv.y, iv.z, iv.w};
    float2 acc = make_float2(0.0f, 0.0f);
#pragma unroll
    for (int l = 0; l < 4; ++l) {
        float2 v = *(const float2*)(features + (size_t)idx[l] * OUT_DIM + 2 * lane);
        float2 f = *(const float2*)(frv + l * 64 + 2 * lane);
        acc.x += v.x * f.x - v.y * f.y;
        acc.y += v.x * f.y + v.y * f.x;
    }
    acc.x *= 0.25f; acc.y *= 0.25f;
    *(float2*)(hid + (size_t)e * OUT_DIM + 2 * lane) = acc;
}

// ---------------------------------------------------------------------------
// Kernel 3: logits via V_WMMA_F32_16X16X4_F32.
// One wave per 16-edge tile: A(16x128)=[center|hidden], B(128x16)=[W1;W2]^T
// (cols 8..15 zero). 32 K-chunks of 4. Then LeakyReLU + store + atomicMax.
// ---------------------------------------------------------------------------
__global__ void __launch_bounds__(32)
logits_kernel(const float* __restrict__ features,
              const int* __restrict__ emi,
              const int* __restrict__ edst,
              const float* __restrict__ attn1_w,
              const float* __restrict__ attn2,
              const float* __restrict__ hid,
              float* __restrict__ aex,
              int* __restrict__ amaxOrd, int E) {
    __shared__ float As[16 * KDIM];   // A[m][k]
    __shared__ float Bs[KDIM * 16];   // B[k][n]
    const int L = threadIdx.x;        // lane 0..31
    const int Lm = L & 15;
    const int tile = blockIdx.x;

    // Stage B = [attn1_w ; attn2]^T, columns >= NUM_HEADS zeroed.
    for (int t = L; t < KDIM * 16; t += 32) {
        int k = t >> 4, n = t & 15;
        float w = 0.0f;
        if (n < NUM_HEADS)
            w = (k < OUT_DIM) ? attn1_w[n * OUT_DIM + k]
                              : attn2[n * OUT_DIM + (k - OUT_DIM)];
        Bs[t] = w;
    }
    // Stage A rows: one row (128 floats) = 32 lanes x float4 per iteration.
    for (int m = 0; m < 16; ++m) {
        int e = tile * 16 + m;
        int k4 = L * 4;
        float4 v = make_float4(0.f, 0.f, 0.f, 0.f);
        if (e < E) {
            if (k4 < OUT_DIM) {
                int cidx = emi[(size_t)e * 4 + 3];
                v = *(const float4*)(features + (size_t)cidx * OUT_DIM + k4);
            } else {
                v = *(const float4*)(hid + (size_t)e * OUT_DIM + (k4 - OUT_DIM));
            }
        }
        *(float4*)(As + m * KDIM + k4) = v;
    }
    __syncthreads();

    // K loop: per ISA f32 layouts, lane half selects K (+0 or +2) within chunk.
    v8f c = {};
    const int koff = (L < 16) ? 0 : 2;
#pragma unroll
    for (int kc = 0; kc < KDIM / 4; ++kc) {
        int kb = kc * 4 + koff;
        v2f av, bv;
        av.x = As[Lm * KDIM + kb];
        av.y = As[Lm * KDIM + kb + 1];
        bv.x = Bs[kb * 16 + Lm];
        bv.y = Bs[(kb + 1) * 16 + Lm];
        c = __builtin_amdgcn_wmma_f32_16x16x4_f32(false, av, false, bv,
                                                  (short)0, c, false, false);
    }

    // D layout: VGPR v -> row m=v (lanes 0-15) / m=v+8 (lanes 16-31), n=Lm.
#pragma unroll
    for (int v = 0; v < 8; ++v) {
        int m = (L < 16) ? v : (v + 8);
        int n = Lm;
        if (n < NUM_HEADS) {
            int e = tile * 16 + m;
            if (e < E) {
                float aval = c[v];
                aval = (aval > 0.0f) ? aval : (ALPHA_C * aval);
                aex[(size_t)e * NUM_HEADS + n] = aval;
                int dst = edst[e];
                atomicMax(&amaxOrd[(size_t)dst * NUM_HEADS + n], f2ord(aval));
            }
        }
    }
}

// ---------------------------------------------------------------------------
// Kernel 4: aexp = exp(a - amax[dst]); segment-sum via atomicAdd.
// ---------------------------------------------------------------------------
__global__ void exp_kernel(const int* __restrict__ edst,
                           const int* __restrict__ amaxOrd,
                           float* __restrict__ aex,
                           float* __restrict__ asum, int E) {
    long long i = (long long)blockIdx.x * blockDim.x + threadIdx.x;
    if (i >= (long long)E * NUM_HEADS) return;
    int e = (int)(i >> 3), h = (int)(i & 7);
    int dst = edst[e];
    float m = ord2f(amaxOrd[(size_t)dst * NUM_HEADS + h]);
    float ex = expf(aex[i] - m);
    aex[i] = ex;
    atomicAdd(&asum[(size_t)dst * NUM_HEADS + h], ex);
}

// ---------------------------------------------------------------------------
// Kernel 5: scatter h[dst,h,:] += (aexp/asum[dst]) * hidden. Wave per edge.
// ---------------------------------------------------------------------------
__global__ void scatter_kernel(const int* __restrict__ edst,
                               const float* __restrict__ hid,
                               const float* __restrict__ aex,
                               const float* __restrict__ asum,
                               float* __restrict__ out, int E) {
    int lane = threadIdx.x & 31;
    int e = blockIdx.x * 8 + (threadIdx.x >> 5);
    if (e >= E) return;
    int dst = edst[e];
    float2 hv = *(const float2*)(hid + (size_t)e * OUT_DIM + 2 * lane);
    float* op = out + (size_t)dst * (NUM_HEADS * OUT_DIM) + 2 * lane;
#pragma unroll
    for (int h = 0; h < NUM_HEADS; ++h) {
        float att = aex[(size_t)e * NUM_HEADS + h] /
                    asum[(size_t)dst * NUM_HEADS + h];
        atomicAdd(op + h * OUT_DIM + 0, att * hv.x);
        atomicAdd(op + h * OUT_DIM + 1, att * hv.y);
    }
}

// ---------------------------------------------------------------------------
extern "C" void kernel_launch(void* const* d_in, const int* in_sizes, int n_in,
                              void* d_out, int out_size, void* d_ws, size_t ws_size,
                              hipStream_t stream) {
    const float* features = (const float*)d_in[0];
    const float* r_vec    = (const float*)d_in[1];
    const float* attn1_w  = (const float*)d_in[2];
    const float* attn2    = (const float*)d_in[3];
    const int*   emi      = (const int*)d_in[4];
    const int*   edst     = (const int*)d_in[5];
    float* out = (float*)d_out;

    const int N = in_sizes[0] / OUT_DIM;
    const int E = in_sizes[5];

    // workspace layout (floats): frv | hid | aexp | amaxOrd(int) | asum
    float* ws   = (float*)d_ws;
    float* frv  = ws;                               // 256
    float* hid  = frv + 256;                        // E*64
    float* aex  = hid + (size_t)E * OUT_DIM;        // E*8
    int*   amaxOrd = (int*)(aex + (size_t)E * NUM_HEADS);  // N*8
    float* asum = (float*)amaxOrd + (size_t)N * NUM_HEADS; // N*8

    {   // init out / amax / asum
        long long total = (long long)N * NUM_HEADS * OUT_DIM;
        int blocks = (int)((total + 255) / 256);
        init_kernel<<<blocks, 256, 0, stream>>>(out, amaxOrd, asum, N);
    }
    frv_kernel<<<1, 32, 0, stream>>>(r_vec, frv);
    {
        int blocks = (E + 7) / 8;
        hidden_kernel<<<blocks, 256, 0, stream>>>(features, emi, frv, hid, E);
    }
    {
        int blocks = (E + 15) / 16;
        logits_kernel<<<blocks, 32, 0, stream>>>(features, emi, edst, attn1_w,
                                                 attn2, hid, aex, amaxOrd, E);
    }
    {
        long long total = (long long)E * NUM_HEADS;
        int blocks = (int)((total + 255) / 256);
        exp_kernel<<<blocks, 256, 0, stream>>>(edst, amaxOrd, aex, asum, E);
    }
    {
        int blocks = (E + 7) / 8;
        scatter_kernel<<<blocks, 256, 0, stream>>>(edst, hid, aex, asum, out, E);
    }
}